// EccPool_8512625181180
// MI455X (gfx1250) — compile-verified
//
#include <hip/hip_runtime.h>

// EccPool for MI455X (gfx1250): each output pixel is a single k x k box-average,
// k chosen by the pixel's ring (disjoint annuli). Strategy:
//   - per 32x32-output tile: input region -> LDS. Interior tiles use the
//     Tensor Data Mover (one tensor_load_to_lds DMA of the 106x112 tile,
//     TENSORcnt-tracked); boundary tiles use masked scalar loads (zero pad).
//   - column prefix sums in LDS (vertical k-sum -> 1 subtract)
//   - horizontal box-sum as Out(16x16) = VS(16xW) x Sel(Wx16) on the WMMA pipe,
//     chunked with V_WMMA_F32_16X16X4_F32 (f32 end-to-end, frees VALU)
// Memory-bound target: ~320 MB traffic @ 23.3 TB/s ~= 14 us floor.

typedef __attribute__((ext_vector_type(2))) float v2f;
typedef __attribute__((ext_vector_type(8))) float v8f;
typedef __attribute__((ext_vector_type(4))) unsigned v4u;
typedef __attribute__((ext_vector_type(8))) unsigned v8u;

#define IN_DIM  512
#define OUT_DIM 256

// LDS column-prefix-sum array: rows 0..106 (row 0 = zero), padded stride 112.
#define CROWS 107
#define CSTR  112

__constant__ int KTAB[14] = {2,5,8,12,15,18,21,24,28,31,34,37,40,44};

__device__ __forceinline__ int ring_of(int oy, int ox) {
  int dy = oy - 128, dx = ox - 128;
  int r2 = dy * dy + dx * dx;
#pragma unroll
  for (int i = 0; i < 13; ++i) {
    int R = 30 + 8 * i;           // disk radii 30,38,...,126 (strict <)
    if (r2 < R * R) return i;
  }
  return 13;                      // outermost: complement of disk(126)
}

__launch_bounds__(128)
__global__ void eccpool_kernel(const float* __restrict__ x, float* __restrict__ out) {
  __shared__ float cum[CROWS * CSTR];   // ~47 KB

  const int c   = blockIdx.z;
  const int ox0 = blockIdx.x * 32;      // output tile origin
  const int oy0 = blockIdx.y * 32;
  const int tid = threadIdx.x;
  const int wave = tid >> 5;

  // input region origin (max halo: p<=21 left/top, k-1-p<=22 right/bottom)
  const int yo = oy0 * 2 - 21;
  const int xo = ox0 * 2 - 21;

  const float* __restrict__ xc = x + (size_t)c * IN_DIM * IN_DIM;

  // Interior tiles: whole 106x112 region is in-bounds -> TDM DMA, no padding.
  const bool interior = (blockIdx.x >= 1) & (blockIdx.x <= 6) &
                        (blockIdx.y >= 1) & (blockIdx.y <= 6);

  if (interior) {
    // ---- Stage 1a: zero row 0; DMA rows 1..106 via the Tensor Data Mover ----
    if (tid < CSTR) cum[tid] = 0.f;
    if (wave == 0) {
      unsigned       ldsa = (unsigned)(uintptr_t)&cum[CSTR];   // LDS byte addr, row 1
      unsigned long long ga =
          (unsigned long long)(uintptr_t)(xc + (size_t)yo * IN_DIM + xo);
      // D# group 0: count=1 | lds_addr | global_addr[56:0] | type=2
      v4u g0;
      g0[0] = 1u;
      g0[1] = ldsa;
      g0[2] = (unsigned)(ga & 0xFFFFFFFFu);
      g0[3] = ((unsigned)(ga >> 32) & 0x01FFFFFFu) | (2u << 30);
      // D# group 1: data_size=4B; tensor_dim0/1 = 2^20 (no clipping needed);
      //             tile = 112 x 106; tensor_dim0_stride = 512 elements.
      v8u g1;
      g1[0] = 2u << 16;                 // [17:16] data_size = 2 (4 bytes)
      g1[1] = 0u;                       // abar=0; tensor_dim0[15:0] = 0
      g1[2] = 0x10u;                    // tensor_dim0[31:16]=0x10; tensor_dim1[15:0]=0
      g1[3] = 0x10u | (112u << 16);     // tensor_dim1[31:16]=0x10; tile_dim0=112
      g1[4] = 106u;                     // tile_dim1=106; tile_dim2=0
      g1[5] = 512u;                     // tensor_dim0_stride[31:0]
      g1[6] = 0u;
      g1[7] = 0u;
      asm volatile("tensor_load_to_lds %0, %1" :: "s"(g0), "s"(g1) : "memory");
      __builtin_amdgcn_s_wait_tensorcnt((unsigned short)0);
    }
  } else {
    // ---- Stage 1b (boundary): masked scalar loads with zero padding ----
    for (int idx = tid; idx < CROWS * CSTR; idx += 128) {
      int ly = idx / CSTR;
      int lx = idx - ly * CSTR;
      float v = 0.f;
      if (ly >= 1) {
        int gy = yo + ly - 1;
        int gx = xo + lx;
        if ((unsigned)gy < (unsigned)IN_DIM && (unsigned)gx < (unsigned)IN_DIM)
          v = xc[gy * IN_DIM + gx];
      }
      cum[idx] = v;
    }
  }
  __syncthreads();

  // ---- Stage 2: column prefix sums: cum[y][x] = sum of input rows yo..yo+y-1 ----
  if (tid < CSTR) {
    float s = 0.f;
    for (int y = 1; y < CROWS; ++y) {
      s += cum[y * CSTR + tid];
      cum[y * CSTR + tid] = s;
    }
  }
  __syncthreads();

  // ---- Stage 3: one wave per 16x16 quadrant; horizontal reduce on WMMA pipe ----
  const int lane = tid & 31;
  const int qy = wave >> 1, qx = wave & 1;
  const int hi = lane >> 4;       // lane half: selects K pair (A/B) and row half (C/D)
  const int n  = lane & 15;       // B/D column, and A row M

  // per-lane ring ids of its 8 C/D elements (VGPR j -> row M = j + 8*hi, col N = n)
  unsigned char ring8[8];
  unsigned present = 0;
#pragma unroll
  for (int j = 0; j < 8; ++j) {
    int oy = oy0 + 16 * qy + j + 8 * hi;
    int ox = ox0 + 16 * qx + n;
    ring8[j] = (unsigned char)ring_of(oy, ox);
    present |= 1u << ring8[j];
  }
  // wave-wide OR of rings present in this quadrant (uniform branch -> EXEC all-1 at WMMA)
#pragma unroll
  for (int off = 16; off; off >>= 1)
    present |= (unsigned)__shfl_xor((int)present, off, 32);

  float res[8];
#pragma unroll
  for (int j = 0; j < 8; ++j) res[j] = 0.f;

  for (int i = 0; i < 14; ++i) {
    if (!((present >> i) & 1u)) continue;
    const int k  = KTAB[i];
    const int p  = (k - 1) >> 1;
    const int Wk = 30 + k;                    // x-extent touched by 16 output cols
    const int lx0 = 32 * qx + 21 - p;         // LDS col of first window column
    const int m   = n;                        // A row M = lane&15
    const int ly1 = 32 * qy + 2 * m + 21 - p; // prefix-sum row indices
    const int ly2 = ly1 + k;                  // (<= 106 by construction)
    const int chunks = (Wk + 3) >> 2;

    v8f acc = {};
    for (int j = 0; j < chunks; ++j) {
      // K=4 slice covers LDS cols xj..xj+3; this lane supplies K = 2*hi, 2*hi+1
      int xA = lx0 + 4 * j + 2 * hi;
      v2f a, b;
      // A[m][K]: vertical k-row sum at column xA (one subtract via prefix sums)
      a.x = cum[ly2 * CSTR + xA]     - cum[ly1 * CSTR + xA];
      a.y = cum[ly2 * CSTR + xA + 1] - cum[ly1 * CSTR + xA + 1];
      // B[K][n]: 1 iff column belongs to output-col n's window [lx0+2n, lx0+2n+k)
      int w0 = xA     - lx0 - 2 * n;
      int w1 = xA + 1 - lx0 - 2 * n;
      b.x = ((unsigned)w0 < (unsigned)k) ? 1.f : 0.f;
      b.y = ((unsigned)w1 < (unsigned)k) ? 1.f : 0.f;
      // D = A x B + C on the matrix pipe
      acc = __builtin_amdgcn_wmma_f32_16x16x4_f32(
          /*neg_a=*/false, a, /*neg_b=*/false, b,
          /*c_mod=*/(short)0, acc, /*reuse_a=*/false, /*reuse_b=*/false);
    }

    const float scale = 1.f / (float)(k * k);   // count_include_pad divisor
#pragma unroll
    for (int j = 0; j < 8; ++j)
      if (ring8[j] == (unsigned char)i) res[j] = acc[j] * scale;
  }

  // ---- store: C/D layout -> row M = j + 8*hi, col N = n ----
#pragma unroll
  for (int j = 0; j < 8; ++j) {
    int oy = oy0 + 16 * qy + j + 8 * hi;
    int ox = ox0 + 16 * qx + n;
    out[((size_t)c * OUT_DIM + oy) * OUT_DIM + ox] = res[j];
  }
}

extern "C" void kernel_launch(void* const* d_in, const int* in_sizes, int n_in,
                              void* d_out, int out_size, void* d_ws, size_t ws_size,
                              hipStream_t stream) {
  (void)in_sizes; (void)n_in; (void)d_ws; (void)ws_size; (void)out_size;
  const float* xin = (const float*)d_in[0];
  float* o = (float*)d_out;
  dim3 grid(OUT_DIM / 32, OUT_DIM / 32, 256);   // 8 x 8 tiles x 256 channels
  dim3 block(128);                               // 4 waves -> 4 quadrants
  eccpool_kernel<<<grid, block, 0, stream>>>(xin, o);
}